// SE3ClassEmbedings_90254442758489
// MI455X (gfx1250) — compile-verified
//
#include <hip/hip_runtime.h>

// ---------------------------------------------------------------------------
// SE3 class embeddings: two chained complex GEMMs on fp32 data, mapped to
// CDNA5 V_WMMA_F32_16X16X4_F32 (wave32 WMMA).
//
//   stage1: mcT[b][n][ch] = sum_t w_l[e,t] * c_l[b,m,t,n]   (ch = off_l + m*E + e)
//   stage2: out[b][n][s]  = sum_ch conj(mcT[b][n][ch]) * x[b][ch][s]
//
// Stage 2: 8-wave workgroups, block tile 128(class) x 64(spatial).
// Double-buffered planar LDS x-tiles; GLOBAL_LOAD_ASYNC_TO_LDS_B32 fill for
// chunk c+1 overlaps the 128-WMMA compute of chunk c (ASYNCcnt pipeline).
// ---------------------------------------------------------------------------

typedef float v2f __attribute__((ext_vector_type(2)));
typedef float v8f __attribute__((ext_vector_type(8)));

#define WMMA_F32(A, B, C) \
  __builtin_amdgcn_wmma_f32_16x16x4_f32(false, (A), false, (B), (short)0, (C), false, false)

#if defined(__has_builtin)
#if __has_builtin(__builtin_amdgcn_global_load_async_to_lds_b32)
#define USE_ASYNC_LDS 1
#endif
#endif
#ifndef USE_ASYNC_LDS
#define USE_ASYNC_LDS 0
#endif

// Builtin signature (from hipcc diagnostic): (int* gaddr, int* lds, int, int)
#define ASYNC_B32(gp, lp) \
  __builtin_amdgcn_global_load_async_to_lds_b32((int*)(gp), (int*)(lp), 0, 0)

constexpr int BATCH = 4;
constexpr int E     = 128;
constexpr int T     = 256;
constexpr int NCLS  = 512;
constexpr int S     = 4096;   // 16*16*16
constexpr int CT    = 2048;   // sum over degrees of (2l+1)*E

constexpr int KC      = 32;          // stage-2 K chunk (divides every degree segment)
constexpr int NCHUNK  = CT / KC;     // 64
constexpr int LDS_LDW = 72;          // padded row stride (floats): 2 rows = 16 mod 64
                                     // -> half-wave bank sets disjoint for frag reads

// ---------------------------------------------------------------------------
// Stage 1: per-(b, l, m) complex GEMM  [NCLS x T] @ [T x E] -> mcT planar.
//   A[row=n][t] = c_l[b, m, t, n]   (complex, interleaved)
//   B[t][col=e] = w_l[e, t]         (complex, interleaved)
// One wave per 16x16 complex output tile.
// ---------------------------------------------------------------------------
__global__ void __launch_bounds__(32)
se3_stage1(const float* __restrict__ c0, const float* __restrict__ c1,
           const float* __restrict__ c2, const float* __restrict__ c3,
           const float* __restrict__ w0, const float* __restrict__ w1,
           const float* __restrict__ w2, const float* __restrict__ w3,
           float* __restrict__ mcT_re, float* __restrict__ mcT_im) {
  const int lane = threadIdx.x & 31;
  const int li   = lane & 15;
  const int half = lane >> 4;

  const int e0 = blockIdx.x * 16;   // E tile
  const int n0 = blockIdx.y * 16;   // class tile
  const int bz = blockIdx.z;        // flattened (b, lm)
  const int b  = bz >> 4;
  const int lm = bz & 15;

  int l, m;
  if (lm == 0)      { l = 0; m = 0;      }
  else if (lm < 4)  { l = 1; m = lm - 1; }
  else if (lm < 9)  { l = 2; m = lm - 4; }
  else              { l = 3; m = lm - 9; }

  const float* cp; const float* wp; int ml; int ch_off;
  switch (l) {
    case 0:  cp = c0; wp = w0; ml = 1; ch_off = 0;    break;
    case 1:  cp = c1; wp = w1; ml = 3; ch_off = 128;  break;
    case 2:  cp = c2; wp = w2; ml = 5; ch_off = 512;  break;
    default: cp = c3; wp = w3; ml = 7; ch_off = 1152; break;
  }
  const int ch0 = ch_off + m * E;

  // c_l flat complex index: ((b*ml + m)*T + t)*NCLS + n
  const size_t cbase = (size_t)(b * ml + m) * T * NCLS;

  const int row = n0 + li;   // class index (A rows, lanes 0..15)
  const int col = e0 + li;   // e index     (B cols, lanes 0..15)

  v8f accR = {}; v8f accI = {};

  for (int k = 0; k < T; k += 4) {
    const int t1 = k + 2 * half;           // K for VGPR0; VGPR1 = t1+1
    v2f ca = *(const v2f*)(cp + 2 * (cbase + (size_t)t1 * NCLS + row));
    v2f cb = *(const v2f*)(cp + 2 * (cbase + (size_t)(t1 + 1) * NCLS + row));
    float4 wv = *(const float4*)(wp + 2 * ((size_t)col * T + t1));

    v2f aR = {ca.x, cb.x}, aI = {ca.y, cb.y};
    v2f bR = {wv.x, wv.z}, bI = {wv.y, wv.w};
    v2f nAI = {-aI.x, -aI.y};

    // (aR + i aI)(bR + i bI): re += aR*bR - aI*bI ; im += aR*bI + aI*bR
    accR = WMMA_F32(aR,  bR, accR);
    accR = WMMA_F32(nAI, bI, accR);
    accI = WMMA_F32(aR,  bI, accI);
    accI = WMMA_F32(aI,  bR, accI);
  }

#pragma unroll
  for (int r = 0; r < 8; ++r) {
    const int ro = n0 + r + 8 * half;
    const size_t o = ((size_t)b * NCLS + ro) * CT + (ch0 + e0 + li);
    mcT_re[o] = accR[r];
    mcT_im[o] = accI[r];
  }
}

// ---------------------------------------------------------------------------
// Stage 2 helpers
// ---------------------------------------------------------------------------
// Pointer to x element (b, global channel ch, spatial col) — complex base.
__device__ __forceinline__ const float*
x_chunk_ptr(const float* x0, const float* x1, const float* x2, const float* x3,
            int b, int ch, int col) {
  const float* xp; int ch_off, ml;
  if (ch < 128)       { xp = x0; ch_off = 0;    ml = 1; }
  else if (ch < 512)  { xp = x1; ch_off = 128;  ml = 3; }
  else if (ch < 1152) { xp = x2; ch_off = 512;  ml = 5; }
  else                { xp = x3; ch_off = 1152; ml = 7; }
  const size_t xb = (size_t)b * (ml * E) * S;
  return xp + 2 * (xb + (size_t)(ch - ch_off) * S + col);
}

// ---------------------------------------------------------------------------
// Stage 2: per batch, complex GEMM  conj(A)[512 x 2048] @ X[2048 x 4096].
//   A[row=n][k=ch] = mcT[b][n][ch]        (planar, K-contiguous, L2-resident)
//   B[k=ch][col=s] = x_l[b][ch_local][s]  (complex, interleaved, streamed)
// ---------------------------------------------------------------------------
__global__ void __launch_bounds__(256)
se3_stage2(const float* __restrict__ x0, const float* __restrict__ x1,
           const float* __restrict__ x2, const float* __restrict__ x3,
           const float* __restrict__ mcT_re, const float* __restrict__ mcT_im,
           float* __restrict__ out) {
  __shared__ float sRe[2][KC * LDS_LDW];
  __shared__ float sIm[2][KC * LDS_LDW];

  const int t    = threadIdx.x;
  const int lane = t & 31;
  const int wave = t >> 5;
  const int li   = lane & 15;
  const int half = lane >> 4;

  const int s0   = blockIdx.x * 64;          // spatial tile (shared by block)
  const int nblk = blockIdx.y * 128;         // class tile base (128 per block)
  const int b    = blockIdx.z;

  const int row = nblk + wave * 16 + li;     // this wave's A rows
  const size_t abase = ((size_t)b * NCLS + row) * CT;

  // fill mapping: thread t, rep i -> (row fr + 4i, col fc)
  const int fr = t >> 6;                     // 0..3
  const int fc = t & 63;                     // 0..63

  v8f accR[4] = {{}, {}, {}, {}};
  v8f accI[4] = {{}, {}, {}, {}};

  // ---- prologue: fill chunk 0 into buffer 0 -------------------------------
  {
    const float* gp = x_chunk_ptr(x0, x1, x2, x3, b, 0, s0 + fc);
#pragma unroll
    for (int i = 0; i < 8; ++i) {
      const int r = fr + 4 * i;
      const float* g = gp + 2 * (size_t)r * S;
#if USE_ASYNC_LDS
      ASYNC_B32(g,     &sRe[0][r * LDS_LDW + fc]);
      ASYNC_B32(g + 1, &sIm[0][r * LDS_LDW + fc]);
#else
      v2f xv = *(const v2f*)g;
      sRe[0][r * LDS_LDW + fc] = xv.x;
      sIm[0][r * LDS_LDW + fc] = xv.y;
#endif
    }
#if USE_ASYNC_LDS
    asm volatile("s_wait_asynccnt 0x0" ::: "memory");
#endif
    __syncthreads();
  }

  // ---- steady state: fill(c+1) overlaps compute(c) ------------------------
  int p = 0;
  for (int c = 0; c < NCHUNK; ++c, p ^= 1) {
    const bool more = (c + 1 < NCHUNK);

    // issue next chunk's fill (async: straight to LDS; manual: into regs)
    v2f stage[8];
    if (more) {
      const float* gnext = x_chunk_ptr(x0, x1, x2, x3, b, (c + 1) * KC, s0 + fc);
#pragma unroll
      for (int i = 0; i < 8; ++i) {
        const int r = fr + 4 * i;
        const float* g = gnext + 2 * (size_t)r * S;
#if USE_ASYNC_LDS
        ASYNC_B32(g,     &sRe[p ^ 1][r * LDS_LDW + fc]);
        ASYNC_B32(g + 1, &sIm[p ^ 1][r * LDS_LDW + fc]);
#else
        stage[i] = *(const v2f*)g;
#endif
        // stream the chunk after next HBM -> L2 (speculative hint)
        __builtin_prefetch(g + 2 * (size_t)KC * S, 0, 1);
      }
    }

    // compute chunk c from buffer p (A uses global channel = c*KC + kq)
    // Fully unrolled: 8 A-load pairs + 64 LDS frag reads scheduled across
    // 128 WMMAs in one straight-line body.
    const int chb = c * KC;
#pragma unroll
    for (int kq = 0; kq < KC; kq += 4) {
      const int k1 = kq + 2 * half;
      v2f aR = *(const v2f*)(mcT_re + abase + chb + k1);
      v2f aI = *(const v2f*)(mcT_im + abase + chb + k1);
      v2f nAI = {-aI.x, -aI.y};

#pragma unroll
      for (int j = 0; j < 4; ++j) {
        const int cix = 16 * j + li;
        v2f bR = { sRe[p][k1 * LDS_LDW + cix], sRe[p][(k1 + 1) * LDS_LDW + cix] };
        v2f bI = { sIm[p][k1 * LDS_LDW + cix], sIm[p][(k1 + 1) * LDS_LDW + cix] };

        // conj(A)*B: re += aR*bR + aI*bI ; im += aR*bI - aI*bR
        accR[j] = WMMA_F32(aR,  bR, accR[j]);
        accR[j] = WMMA_F32(aI,  bI, accR[j]);
        accI[j] = WMMA_F32(aR,  bI, accI[j]);
        accI[j] = WMMA_F32(nAI, bR, accI[j]);
      }
    }

#if USE_ASYNC_LDS
    asm volatile("s_wait_asynccnt 0x0" ::: "memory");
#else
    if (more) {
#pragma unroll
      for (int i = 0; i < 8; ++i) {
        const int r = fr + 4 * i;
        sRe[p ^ 1][r * LDS_LDW + fc] = stage[i].x;
        sIm[p ^ 1][r * LDS_LDW + fc] = stage[i].y;
      }
    }
#endif
    __syncthreads();
  }

  // ---- epilogue: interleaved complex store, coalesced per half-wave -------
#pragma unroll
  for (int j = 0; j < 4; ++j) {
#pragma unroll
    for (int r = 0; r < 8; ++r) {
      const int ro  = nblk + wave * 16 + r + 8 * half;
      const int col = s0 + 16 * j + li;
      const size_t o = (((size_t)b * NCLS + ro) * S + col) * 2;
      *(v2f*)(out + o) = (v2f){accR[j][r], accI[j][r]};
    }
  }
}

// ---------------------------------------------------------------------------
// Launch. Input order (setup_inputs insertion order):
//   0:x0 1:c0 2:w0 3:x1 4:c1 5:w1 6:x2 7:c2 8:w2 9:x3 10:c3 11:w3
// complex64 -> interleaved float pairs. Workspace: mcT_re | mcT_im (32 MB).
// ---------------------------------------------------------------------------
extern "C" void kernel_launch(void* const* d_in, const int* in_sizes, int n_in,
                              void* d_out, int out_size, void* d_ws, size_t ws_size,
                              hipStream_t stream) {
  const float* x0 = (const float*)d_in[0];
  const float* c0 = (const float*)d_in[1];
  const float* w0 = (const float*)d_in[2];
  const float* x1 = (const float*)d_in[3];
  const float* c1 = (const float*)d_in[4];
  const float* w1 = (const float*)d_in[5];
  const float* x2 = (const float*)d_in[6];
  const float* c2 = (const float*)d_in[7];
  const float* w2 = (const float*)d_in[8];
  const float* x3 = (const float*)d_in[9];
  const float* c3 = (const float*)d_in[10];
  const float* w3 = (const float*)d_in[11];

  float* mcT_re = (float*)d_ws;
  float* mcT_im = mcT_re + (size_t)BATCH * NCLS * CT;

  // Stage 1: grid = (E/16, NCLS/16, B * 16 (b,l,m) pairs), 1 wave each.
  se3_stage1<<<dim3(E / 16, NCLS / 16, BATCH * 16), 32, 0, stream>>>(
      c0, c1, c2, c3, w0, w1, w2, w3, mcT_re, mcT_im);

  // Stage 2: grid = (S/64, NCLS/128, B), 256 threads (8 waves) each.
  se3_stage2<<<dim3(S / 64, NCLS / 128, BATCH), 256, 0, stream>>>(
      x0, x1, x2, x3, mcT_re, mcT_im, (float*)d_out);
}